// CrossPool_53841710023024
// MI455X (gfx1250) — compile-verified
//
#include <hip/hip_runtime.h>
#include <hip/hip_bf16.h>
#include <math.h>

// CrossPool (AXIS=2, POOL_MODE='max', SPATIAL_SCALE=1.0) for MI455X / gfx1250.
// input : (N, C, H, W) fp32
// rois  : (N, H, W, 6) fp32  -> [x1, x2, yc, xc, y1, y2]
// out   : (N, 2C, H, W) fp32 : out[:, 0:C]  = horiz max over row yc, cols [x1..x2]
//                              out[:, C:2C] = vert  max over col xc, rows [y1..y2]
//
// One block per (n,h,w) query. The channel-replicated gathers are TDM-shaped
// tile loads -> tensor_load_to_lds DMAs the tiles into LDS, double-buffered so
// the TDM prefetches chunk k+1 while the 8 waves reduce chunk k from LDS
// (TENSORcnt is in-order per wave: wait<=2 ==> all but the newest pair done).

#define TDM_PATH (__has_builtin(__builtin_amdgcn_tensor_load_to_lds) && \
                  __has_builtin(__builtin_amdgcn_s_wait_tensorcnt))

__device__ __forceinline__ int clampi(int v, int lo, int hi) {
    return v < lo ? lo : (v > hi ? hi : v);
}

#if TDM_PATH

typedef unsigned int v4u __attribute__((ext_vector_type(4)));
typedef int          v8i __attribute__((ext_vector_type(8)));
typedef int          v4i __attribute__((ext_vector_type(4)));

// Issue one TDM tile load: 3D tile (tile_d0 x tile_d1 x tile_d2) of fp32,
// element strides (1, stride0, stride1) in fp32 units, packed contiguously
// into LDS at byte offset lds_off. Tensor dims == tile dims (clamping already
// guarantees in-bounds, so no OOB zero-fill can occur).
// D# bit layout per CDNA5 ISA.
// 6-arg builtin (groups 0..3 + aux octet + cpol); aux zero-filled.
__device__ __forceinline__ void tdm_load_tile_f32(
    const float* gptr, unsigned lds_off,
    unsigned tile_d0, unsigned tile_d1, unsigned tile_d2,
    unsigned long long stride0, unsigned long long stride1)
{
    const unsigned long long ga = (unsigned long long)(uintptr_t)gptr;

    // ---- D# group 0 (4 dwords)
    const v4u g0 = {
        0x1u,                                        // count=1, user-mode load
        lds_off,                                     // lds_addr (bytes)
        (unsigned)(ga & 0xFFFFFFFFu),                // global_addr[31:0]
        (unsigned)((ga >> 32) & 0x01FFFFFFu)         // global_addr[56:32]
            | 0x80000000u                            // type=2 ("image")
    };

    // ---- D# group 1 (8 dwords)
    const unsigned td0 = tile_d0, td1 = tile_d1;     // tensor dims = tile dims
    const v8i g1 = {
        (int)(2u << 16),                                              // data_size=4B; mask=0
        (int)((td0 & 0xFFFFu) << 16),                                 // tensor_dim0 lo
        (int)(((td0 >> 16) & 0xFFFFu) | ((td1 & 0xFFFFu) << 16)),     // dim0 hi | dim1 lo
        (int)(((td1 >> 16) & 0xFFFFu) | ((tile_d0 & 0xFFFFu) << 16)), // dim1 hi | tile_dim0
        (int)((tile_d1 & 0xFFFFu) | ((tile_d2 & 0xFFFFu) << 16)),     // tile_dim1 | tile_dim2
        (int)(stride0 & 0xFFFFFFFFu),                                 // dim0_stride lo32
        (int)(((stride0 >> 32) & 0xFFFFu) | ((stride1 & 0xFFFFu) << 16)), // s0 hi | s1 lo
        (int)((stride1 >> 16) & 0xFFFFFFFFu)                          // dim1_stride[47:16]
    };

    // ---- D# group 2: tensor_dim2 = tile_d2; tile_dim3=0 (3D only)
    const v4i g2 = { (int)tile_d2, 0, 0, 0 };
    // ---- D# group 3: unused
    const v4i g3 = { 0, 0, 0, 0 };
    // ---- aux octet (6-arg builtin form): zero-filled
    const v8i gaux = { 0, 0, 0, 0, 0, 0, 0, 0 };

    __builtin_amdgcn_tensor_load_to_lds(g0, g1, g2, g3, gaux, /*cpol=*/0);
}

#define CB      64            // channels per TDM chunk
#define NCHUNK  4             // C / CB (C = 256)
#define TILE_F  (56 * CB)     // max floats per tile (span <= 56)

__global__ __launch_bounds__(256) void cross_pool_max_tdm_kernel(
    const float* __restrict__ inp,
    const float* __restrict__ rois,
    float* __restrict__ out,
    const int N, const int C, const int H, const int W)
{
    // double-buffered tiles: 2 * 2 * 56*64*4 = 56 KB LDS
    __shared__ float hbuf[2][TILE_F];  // horizontal tile: [c_local][spanX] packed
    __shared__ float vbuf[2][TILE_F];  // vertical   tile: [c_local][spanY] packed

    const int q  = blockIdx.x;         // (n*H + h)*W + w
    const int w  = q % W;
    const int nh = q / W;
    const int h  = nh % H;
    const int n  = nh / H;

    // ROI decode (block-uniform -> scalar loads / SALU)
    const float* __restrict__ r = rois + (size_t)q * 6;
    const int x1 = clampi((int)rintf(r[0]), 0, W - 1);
    int       x2 = clampi((int)rintf(r[1]), 0, W - 1); if (x2 < x1) x2 = x1;
    const int yc = clampi((int)rintf(r[2]), 0, H - 1);
    const int xc = clampi((int)rintf(r[3]), 0, W - 1);
    const int y1 = clampi((int)rintf(r[4]), 0, H - 1);
    int       y2 = clampi((int)rintf(r[5]), 0, H - 1); if (y2 < y1) y2 = y1;

    const int spanX = x2 - x1 + 1;
    const int spanY = y2 - y1 + 1;
    const size_t HW = (size_t)H * W;

    const int c_local = threadIdx.x >> 2;  // 0..63 : channel within chunk
    const int part    = threadIdx.x & 3;   // 4 lanes (same wave) per channel

    // issue the two DMA tile gathers for one channel chunk into buffer `buf`
    auto issue_chunk = [&](int chunk, int buf) {
        const float* hsrc = inp + ((size_t)(n * C + chunk) * H + yc) * W + x1;
        tdm_load_tile_f32(hsrc, (unsigned)(uintptr_t)&hbuf[buf][0],
                          (unsigned)spanX, 1u, (unsigned)CB,
                          (unsigned long long)W, (unsigned long long)HW);
        const float* vsrc = inp + ((size_t)(n * C + chunk) * H + y1) * W + xc;
        tdm_load_tile_f32(vsrc, (unsigned)(uintptr_t)&vbuf[buf][0],
                          1u, (unsigned)spanY, (unsigned)CB,
                          (unsigned long long)W, (unsigned long long)HW);
    };

    // prologue: prefetch chunk 0
    if (threadIdx.x < 32) issue_chunk(0, 0);

    for (int k = 0; k < NCHUNK; ++k) {
        const int buf = k & 1;
        if (threadIdx.x < 32) {
            if (k + 1 < NCHUNK) {
                issue_chunk(k + 1, (k + 1) & 1);       // prefetch next chunk
                __builtin_amdgcn_s_wait_tensorcnt(2);  // chunk k's pair is done
            } else {
                __builtin_amdgcn_s_wait_tensorcnt(0);  // drain
            }
        }
        __syncthreads();   // LDS tiles of chunk k visible to all 8 waves

        // ---- reduce chunk k from LDS: 4 lanes per channel, shuffle-combine
        float mh = -INFINITY;
        {
            const float* src = &hbuf[buf][c_local * spanX];
            for (int x = part; x < spanX; x += 4) mh = fmaxf(mh, src[x]);
        }
        float mv = -INFINITY;
        {
            const float* src = &vbuf[buf][c_local * spanY];
            for (int y = part; y < spanY; y += 4) mv = fmaxf(mv, src[y]);
        }
        mh = fmaxf(mh, __shfl_xor(mh, 1, 32));
        mh = fmaxf(mh, __shfl_xor(mh, 2, 32));
        mv = fmaxf(mv, __shfl_xor(mv, 1, 32));
        mv = fmaxf(mv, __shfl_xor(mv, 2, 32));

        if (part == 0) {
            const int c = k * CB + c_local;
            const size_t obase = (((size_t)n * (2 * C) + c) * H + h) * (size_t)W + w;
            out[obase]          = mh;   // channel block [0, C)
            out[obase + C * HW] = mv;   // channel block [C, 2C)
        }
        __syncthreads();   // all readers done before this buffer is re-filled
    }
}

#else  // !TDM_PATH --------------------------------------------------------

__global__ __launch_bounds__(256) void cross_pool_max_tdm_kernel(
    const float* __restrict__ inp,
    const float* __restrict__ rois,
    float* __restrict__ out,
    const int N, const int C, const int H, const int W)
{
    const int q  = blockIdx.x;
    const int w  = q % W;
    const int nh = q / W;
    const int h  = nh % H;
    const int n  = nh / H;

    const float* __restrict__ r = rois + (size_t)q * 6;
    const int x1 = clampi((int)rintf(r[0]), 0, W - 1);
    int       x2 = clampi((int)rintf(r[1]), 0, W - 1); if (x2 < x1) x2 = x1;
    const int yc = clampi((int)rintf(r[2]), 0, H - 1);
    const int xc = clampi((int)rintf(r[3]), 0, W - 1);
    const int y1 = clampi((int)rintf(r[4]), 0, H - 1);
    int       y2 = clampi((int)rintf(r[5]), 0, H - 1); if (y2 < y1) y2 = y1;

    const size_t HW = (size_t)H * W;

    for (int c = threadIdx.x; c < C; c += blockDim.x) {
        const float* __restrict__ chan = inp + ((size_t)(n * C + c)) * HW;

        const float* __restrict__ rowp = chan + (size_t)yc * W;
        __builtin_prefetch(rowp + x1, 0, 0);
        float mh = -INFINITY;
        int x = x1;
        for (; (x & 3) != 0 && x <= x2; ++x) mh = fmaxf(mh, rowp[x]);
        for (; x + 3 <= x2; x += 4) {
            const float4 v = *reinterpret_cast<const float4*>(rowp + x);
            mh = fmaxf(mh, fmaxf(fmaxf(v.x, v.y), fmaxf(v.z, v.w)));
        }
        for (; x <= x2; ++x) mh = fmaxf(mh, rowp[x]);

        const float* __restrict__ colp = chan + xc;
        __builtin_prefetch(colp + (size_t)y1 * W, 0, 0);
        float mv = -INFINITY;
        #pragma unroll 4
        for (int y = y1; y <= y2; ++y) mv = fmaxf(mv, colp[(size_t)y * W]);

        const size_t obase = (((size_t)n * (2 * C) + c) * H + h) * (size_t)W + w;
        out[obase]          = mh;
        out[obase + C * HW] = mv;
    }
}

#endif  // TDM_PATH

extern "C" void kernel_launch(void* const* d_in, const int* in_sizes, int n_in,
                              void* d_out, int out_size, void* d_ws, size_t ws_size,
                              hipStream_t stream) {
    (void)n_in; (void)out_size; (void)d_ws; (void)ws_size;

    const float* inp  = (const float*)d_in[0];   // (N, C, H, W) fp32
    const float* rois = (const float*)d_in[1];   // (N, H, W, 6) fp32
    float* out        = (float*)d_out;           // (N, 2C, H, W) fp32

    const int C = 256, H = 56, W = 56;
    const int N = in_sizes[1] / (H * W * 6);     // rois = N*H*W*6

    const int n_queries = N * H * W;             // one block per (n, h, w)
    cross_pool_max_tdm_kernel<<<dim3(n_queries), dim3(256), 0, stream>>>(
        inp, rois, out, N, C, H, W);
}